// BilinearInteractionLayer_60249801228976
// MI455X (gfx1250) — compile-verified
//
#include <hip/hip_runtime.h>

typedef __attribute__((ext_vector_type(2))) float v2f;
typedef __attribute__((ext_vector_type(8))) float v8f;

#define NUM_FIELDS 32
#define EMB        64
#define NPAIRS     496
#define BTILE      128
#define LDSTRIDE   68   // 64 + 4: float4-aligned, conflict-free column access

__global__ __launch_bounds__(256) void bilinear_wmma_kernel(
    const float* __restrict__ femb,   // [2048, 32, 64]
    const float* __restrict__ W,      // [496, 64, 64]  (p, e, d)
    float* __restrict__ out)          // [2048, 496, 64]
{
  __shared__ float smem[(EMB + BTILE) * LDSTRIDE];
  float* ldsW = smem;                    // [64][LDSTRIDE]  W[p][e][d]
  float* ldsA = smem + EMB * LDSTRIDE;   // [128][LDSTRIDE] v_i tile

  const int p      = blockIdx.x;
  const int b_base = blockIdx.y * BTILE;

  // Decode lexicographic pair index p -> (fi, fj), uniform scalar loop.
  int fi = 0, rem = p;
  while (rem >= (NUM_FIELDS - 1 - fi)) { rem -= (NUM_FIELDS - 1 - fi); ++fi; }
  const int fj = fi + 1 + rem;

  const int tid = threadIdx.x;

  // Stage W[p] (64x64) into LDS, float4 vectorized.
  {
    const float4* src = reinterpret_cast<const float4*>(W + (size_t)p * EMB * EMB);
    #pragma unroll
    for (int it = 0; it < 4; ++it) {
      int idx = tid + it * 256;          // 1024 float4s total
      int e   = idx >> 4;
      int d4  = idx & 15;
      float4 v = src[idx];
      *reinterpret_cast<float4*>(&ldsW[e * LDSTRIDE + d4 * 4]) = v;
    }
  }
  // Stage v_i tile (128 rows x 64) into LDS.
  {
    #pragma unroll
    for (int it = 0; it < 8; ++it) {
      int idx = tid + it * 256;          // 2048 float4s total
      int r   = idx >> 4;
      int d4  = idx & 15;
      const float4* row = reinterpret_cast<const float4*>(
          femb + ((size_t)(b_base + r) * NUM_FIELDS + fi) * EMB);
      *reinterpret_cast<float4*>(&ldsA[r * LDSTRIDE + d4 * 4]) = row[d4];
    }
  }
  __syncthreads();

  const int wave   = tid >> 5;
  const int lane   = tid & 31;
  const int lane15 = lane & 15;
  const int khi    = lane >> 4;           // 0: lanes 0-15, 1: lanes 16-31

  const int e_base = (wave & 3) * 16;     // 4 e-slices of 16
  const int mrow0  = (wave >> 2) * 64;    // 2 wave-rows x 64 batch rows

  // Preload B fragments: B[k][n] = W[p][e=n][d=k] (i.e. W^T slice, K-chunked by 4).
  // ISA 32-bit B layout: VGPR0 -> K = d0 = 4k + 2*khi, VGPR1 -> K = d0 + 1.
  v2f bfrag[16];
  const int n = e_base + lane15;
  #pragma unroll
  for (int k = 0; k < 16; ++k) {
    int d0 = k * 4 + 2 * khi;
    bfrag[k][0] = ldsW[n * LDSTRIDE + d0];
    bfrag[k][1] = ldsW[n * LDSTRIDE + d0 + 1];
  }

  #pragma unroll
  for (int ms = 0; ms < 4; ++ms) {
    const int m0 = mrow0 + ms * 16;
    v8f acc = {0.f, 0.f, 0.f, 0.f, 0.f, 0.f, 0.f, 0.f};
    const float* arow = &ldsA[(m0 + lane15) * LDSTRIDE];
    #pragma unroll
    for (int k = 0; k < 16; ++k) {
      int d0 = k * 4 + 2 * khi;
      v2f a;
      a[0] = arow[d0];
      a[1] = arow[d0 + 1];
      // D(16x16,f32) += A(16x4) * B(4x16), exact fp32 accumulate
      acc = __builtin_amdgcn_wmma_f32_16x16x4_f32(
          /*neg_a=*/false, a, /*neg_b=*/false, bfrag[k],
          /*c_mod=*/(short)0, acc, /*reuse_a=*/false, /*reuse_b=*/false);
    }
    // Epilogue: multiply by v_j and stream out (non-temporal: 260 MB write-once).
    // D layout: VGPR r -> M = r + 8*khi, N = lane15.
    #pragma unroll
    for (int r = 0; r < 8; ++r) {
      int b = b_base + m0 + r + 8 * khi;
      int e = e_base + lane15;
      float vj = femb[((size_t)b * NUM_FIELDS + fj) * EMB + e];
      float res = acc[r] * vj;
      __builtin_nontemporal_store(res, &out[((size_t)b * NPAIRS + p) * EMB + e]);
    }
  }
}

extern "C" void kernel_launch(void* const* d_in, const int* in_sizes, int n_in,
                              void* d_out, int out_size, void* d_ws, size_t ws_size,
                              hipStream_t stream) {
  const float* femb = (const float*)d_in[0];  // 2048*32*64 f32
  const float* W    = (const float*)d_in[1];  // 496*64*64 f32
  float* out        = (float*)d_out;          // 2048*496*64 f32

  dim3 grid(NPAIRS, 2048 / BTILE);            // 496 x 16 workgroups
  dim3 block(256);                            // 8 wave32s
  bilinear_wmma_kernel<<<grid, block, 0, stream>>>(femb, W, out);
}